// GNNLayer_42975442763784
// MI455X (gfx1250) — compile-verified
//
#include <hip/hip_runtime.h>
#include <hip/hip_bf16.h>

typedef __attribute__((ext_vector_type(2))) float v2f;
typedef __attribute__((ext_vector_type(8))) float v8f;

#define NN   32768   // nodes (8*64*64)
#define BB   128     // batch
#define OUTN 256     // output features
#define KCHUNKS 32
#define KC   (NN / KCHUNKS)   // 1024 K per chunk

// ---------------------------------------------------------------------------
// 1) Tiled transpose: x (B, N) row-major  ->  xT (N, B) row-major
//    Makes both the edge gather and the atomic scatter fully coalesced.
// ---------------------------------------------------------------------------
__global__ void __launch_bounds__(1024) transpose_xt(const float* __restrict__ x,
                                                     float* __restrict__ xT) {
    __shared__ float tile[32][33];
    const int n0 = blockIdx.x * 32;
    const int b0 = blockIdx.y * 32;
    const int tx = threadIdx.x, ty = threadIdx.y;
    // coalesced read along N
    tile[ty][tx] = x[(size_t)(b0 + ty) * NN + (n0 + tx)];
    __syncthreads();
    // coalesced write along B
    xT[(size_t)(n0 + ty) * BB + (b0 + tx)] = tile[tx][ty];
}

// ---------------------------------------------------------------------------
// 2) COO scatter-add: h1[r, b] += v * xT[c, b]   (one thread per (edge,b))
//    Consecutive threads share an edge -> coalesced loads & coalesced atomics.
// ---------------------------------------------------------------------------
__global__ void __launch_bounds__(256) scatter_edges(const int* __restrict__ rows,
                                                     const int* __restrict__ cols,
                                                     const float* __restrict__ vals,
                                                     const float* __restrict__ xT,
                                                     float* __restrict__ h1,
                                                     int nE) {
    const int t = blockIdx.x * 256 + threadIdx.x;
    if (t >= nE * BB) return;
    const int e = t >> 7;         // edge index
    const int b = t & (BB - 1);   // batch lane
    const int r = rows[e];
    const int c = cols[e];
    const float v = vals[e];
    atomicAdd(&h1[(size_t)r * BB + b], v * xT[(size_t)c * BB + b]);
}

// ---------------------------------------------------------------------------
// 3) Initialize output with the bias (harness poisons d_out; we must fully
//    rewrite it every call). out is (B, OUT) row-major.
// ---------------------------------------------------------------------------
__global__ void __launch_bounds__(256) init_out_bias(const float* __restrict__ bias,
                                                     float* __restrict__ out) {
    const int i = blockIdx.x * 256 + threadIdx.x;   // i in [0, BB*OUTN)
    out[i] = bias[i & (OUTN - 1)];
}

// ---------------------------------------------------------------------------
// 4) fp32 WMMA GEMM with K-split + atomic reduction:
//    out[b, o] += sum_n W[o, n] * h1[n, b]
//    A-tile = W rows   (16 x 4 f32): lane = M(=o), VGPR0/1 = K{0,1} | K{2,3}
//    B-tile = h1       ( 4 x 16 f32): lane = N(=b), VGPR0/1 = K{0,1} | K{2,3}
//    D-tile (16x16 f32): VGPR j -> M = j (+8 for lanes 16-31), N = lane&15
// ---------------------------------------------------------------------------
__global__ void __launch_bounds__(256) gemm_wmma_f32(const float* __restrict__ h1,
                                                     const float* __restrict__ Wm,
                                                     float* __restrict__ out) {
    const int lane = threadIdx.x & 31;
    const int wid  = blockIdx.x * (blockDim.x >> 5) + (threadIdx.x >> 5);
    const int kch  = wid & (KCHUNKS - 1);   // K chunk
    const int tile = wid >> 5;              // 0..127
    const int ot   = tile & 15;             // o-tile (16 of them)
    const int bt   = tile >> 4;             // b-tile (8 of them)
    const int l    = lane & 15;
    const int half = lane >> 4;             // 0: K={0,1}, 1: K={2,3}

    const int o0 = ot * 16;
    const int b0 = bt * 16;
    const int wrow = (o0 + l) * NN;         // this lane's W row (streams K)

    v8f c = {};
    const int nbase = kch * KC;
    for (int kk = 0; kk < KC; kk += 4) {
        const int k0 = nbase + kk + 2 * half;
        v2f a, b;
        a.x = Wm[wrow + k0];
        a.y = Wm[wrow + k0 + 1];
        b.x = h1[(size_t)k0       * BB + b0 + l];
        b.y = h1[(size_t)(k0 + 1) * BB + b0 + l];
        // D = A x B + C  (fp32 matrix pipe, full precision vs. reference)
        c = __builtin_amdgcn_wmma_f32_16x16x4_f32(
                /*neg_a=*/false, a, /*neg_b=*/false, b,
                /*c_mod=*/(short)0, c, /*reuse_a=*/false, /*reuse_b=*/false);
    }

    // D[M = o0 + j + 8*half][N = b0 + l] -> out[(b0+l)*OUTN + o]
    const int bcol = b0 + l;
#pragma unroll
    for (int j = 0; j < 8; ++j) {
        const int o = o0 + j + 8 * half;
        atomicAdd(&out[(size_t)bcol * OUTN + o], c[j]);
    }
}

// ---------------------------------------------------------------------------
extern "C" void kernel_launch(void* const* d_in, const int* in_sizes, int n_in,
                              void* d_out, int out_size, void* d_ws, size_t ws_size,
                              hipStream_t stream) {
    (void)n_in; (void)out_size; (void)ws_size;

    const float* x    = (const float*)d_in[0];  // (128, 8, 64, 64) fp32
    const int*   rows = (const int*)  d_in[1];
    const int*   cols = (const int*)  d_in[2];
    const float* vals = (const float*)d_in[3];
    const float* Wm   = (const float*)d_in[4];  // (256, 32768) fp32
    const float* bias = (const float*)d_in[5];  // (256,)
    float*       out  = (float*)d_out;          // (128, 256) fp32

    float* xT = (float*)d_ws;                   // (N, B)  16.8 MB
    float* h1 = xT + (size_t)NN * BB;           // (N, B)  16.8 MB

    const int nE = in_sizes[1];                 // COO entry count (~606k)

    // zero the scatter accumulator (graph-capturable memset node)
    hipMemsetAsync(h1, 0, (size_t)NN * BB * sizeof(float), stream);

    // transpose x -> xT
    transpose_xt<<<dim3(NN / 32, BB / 32), dim3(32, 32), 0, stream>>>(x, xT);

    // sparse scatter-add
    const int total = nE * BB;
    scatter_edges<<<(total + 255) / 256, 256, 0, stream>>>(rows, cols, vals, xT, h1, nE);

    // bias init then WMMA GEMM with K-split atomic reduction
    init_out_bias<<<(BB * OUTN) / 256, 256, 0, stream>>>(bias, out);
    gemm_wmma_f32<<<(8 * 16 * KCHUNKS) / 8, 256, 0, stream>>>(h1, Wm, out);
}